// bvp_rr_loss_14448269984180
// MI455X (gfx1250) — compile-verified
//
#include <hip/hip_runtime.h>
#include <math.h>

// ---------------- problem constants ----------------
#define TLEN     9000
#define NPERSEG  256
#define STEP     128
#define KMAX     34          // max welch segments
#define MTMAX    3           // ceil(34/16) row tiles -> 48 padded rows
#define NT       5           // 5 col tiles -> 80 padded band bins
#define LOBIN    31          // first bin with f>=0.12  (31/256)
#define NBAND    72          // bins 31..102 inclusive
#define SEGSTRIDE 260        // f32 row stride (260%64==4 -> conflict-free tile reads)
#define NTHREADS 256
#define PI_OVER_128 0.024543692442291664f   // pi/128, for angle = (f*t mod 256)*pi/128

// ---------------- dynamic LDS layout (bytes) ----------------
// [0,     49920)  xbuf (first 36000 B) then reused as segs f32[48*260]
// [49920, 67920)  iv    f32[4500]
// [67920, 76928)  peaks u16[4500] (+pad)
// [76928, 77952)  cnt   i32[256]
// [77952, 78144)  mean  f32[48]
// [78144, 80704)  psd_part f32[8][80]
// [80704, 81024)  psd   f32[80]
// [81024, 81056)  scalars
#define SMEM_BYTES 81056

typedef float v2f __attribute__((ext_vector_type(2)));
typedef float v8f __attribute__((ext_vector_type(8)));
typedef int   v4i __attribute__((ext_vector_type(4)));

#define AS1 __attribute__((address_space(1)))
#define AS3 __attribute__((address_space(3)))

#if __has_builtin(__builtin_amdgcn_global_load_async_to_lds_b128) && \
    __has_builtin(__builtin_amdgcn_s_wait_asynccnt)
#define USE_ASYNC_LDS 1
#else
#define USE_ASYNC_LDS 0
#endif

__global__ void bvp_rr_block_kernel(const float* __restrict__ gt,
                                    const float* __restrict__ pre,
                                    float* __restrict__ contrib)
{
    extern __shared__ char smem[];
    float*          xbuf  = (float*)(smem);
    float*          segs  = (float*)(smem);              // aliases xbuf (time-disjoint)
    float*          iv    = (float*)(smem + 49920);
    unsigned short* peaks = (unsigned short*)(smem + 67920);
    int*            cnt   = (int*)(smem + 76928);
    float*          meanb = (float*)(smem + 77952);
    float*          psdp  = (float*)(smem + 78144);
    float*          psd   = (float*)(smem + 80704);
    int*            scal  = (int*)(smem + 81024);

    const int tid  = threadIdx.x;
    const int b    = blockIdx.x;
    const int w    = tid >> 5;
    const int lane = tid & 31;
    const float* g = gt + (long)b * TLEN;

    // ---- stage 1: stage trace into LDS (9000 = 2250 * 16B) ----
#if USE_ASYNC_LDS
    // CDNA5 async pipe: memory -> LDS directly, no VGPR round-trip (ASYNCcnt).
    {
        AS1 v4i* gv   = (AS1 v4i*)g;          // global, 16B granules
        AS3 v4i* ldsv = (AS3 v4i*)smem;       // LDS dest, 16B granules
        for (int v = tid; v < TLEN / 4; v += NTHREADS)
            __builtin_amdgcn_global_load_async_to_lds_b128(gv + v, ldsv + v, 0, 0);
        __builtin_amdgcn_s_wait_asynccnt(0);  // this wave's transfers done
    }
#else
    for (int v = tid; v < TLEN / 4; v += NTHREADS)
        ((float4*)xbuf)[v] = ((const float4*)g)[v];
#endif
    __syncthreads();                          // all waves' transfers done

    // ---- stage 2: strict local maxima (normalization skipped: monotone) ----
    int c = 0;
    const int i0 = 1 + tid * 36;
    const int i1 = min(i0 + 36, TLEN - 1);
    for (int i = i0; i < i1; ++i)
        c += (xbuf[i] > xbuf[i - 1]) & (xbuf[i] > xbuf[i + 1]);
    cnt[tid] = c;
    __syncthreads();
    // inclusive Hillis-Steele scan over 256 counts
    for (int off = 1; off < NTHREADS; off <<= 1) {
        int a = cnt[tid];
        int p = (tid >= off) ? cnt[tid - off] : 0;
        __syncthreads();
        cnt[tid] = a + p;
        __syncthreads();
    }
    int base = cnt[tid] - c;   // exclusive offset
    for (int i = i0; i < i1; ++i)
        if ((xbuf[i] > xbuf[i - 1]) & (xbuf[i] > xbuf[i + 1]))
            peaks[base++] = (unsigned short)i;
    if (tid == NTHREADS - 1) scal[0] = cnt[tid];
    __syncthreads();

    const int n = scal[0];
    const int L = n - 1;
    const int nsegs = (L >= NPERSEG) ? min((L - NPERSEG) / STEP + 1, KMAX) : 0;
    const int mtiles = (nsegs + 15) >> 4;      // skip all-zero row tiles

    // ---- stage 3: rr intervals ----
    for (int j = tid; j < L; j += NTHREADS)
        iv[j] = (float)(peaks[j + 1] - peaks[j]) * (1.0f / 30.0f);
    __syncthreads();

    // ---- stage 4: per-segment means (one wave per segment) + zero psd_part ----
    for (int k = w; k < nsegs; k += 8) {
        float s = 0.f;
        for (int j = lane; j < NPERSEG; j += 32) s += iv[k * STEP + j];
        for (int off = 16; off > 0; off >>= 1) s += __shfl_xor(s, off, 32);
        if (lane == 0) meanb[k] = s * (1.0f / NPERSEG);
    }
    for (int e = tid; e < 8 * 80; e += NTHREADS) psdp[e] = 0.f;
    __syncthreads();                            // xbuf now dead; segs may overwrite

    // ---- stage 5: windowed, detrended segment matrix (padded rows = 0) ----
    for (int e = tid; e < MTMAX * 16 * NPERSEG; e += NTHREADS) {
        int r = e >> 8, t = e & 255;
        float val = 0.f;
        if (r < nsegs) {
            float hann = 0.5f - 0.5f * __cosf((float)t * PI_OVER_128); // periodic Hann
            val = (iv[r * STEP + t] - meanb[r]) * hann;
        }
        segs[r * SEGSTRIDE + t] = val;
    }
    __syncthreads();

    // ---- stage 6: band-limited DFT as f32 WMMA GEMMs ----
    // C[k][f] = sum_t segs[k][t]*cos(2*pi*f*t/256) ; S analogous with sin.
    // All scale factors (1/WSS, doubling, 1/nsegs) dropped: positive constants,
    // argmax-invariant.
    for (int task = w; task < mtiles * NT; task += 8) {
        const int mtile = task / NT, ntile = task % NT;
        const int mn    = lane & 15;
        const int khalf = (lane >> 4) << 1;          // lanes<16 -> K{0,1}, else K{2,3}
        const int fbin  = LOBIN + ntile * 16 + mn;   // this lane's B column frequency
        const float* arow = segs + (mtile * 16 + mn) * SEGSTRIDE + khalf;
        v8f accC = {0.f, 0.f, 0.f, 0.f, 0.f, 0.f, 0.f, 0.f};
        v8f accS = {0.f, 0.f, 0.f, 0.f, 0.f, 0.f, 0.f, 0.f};
        for (int ks = 0; ks < NPERSEG / 4; ++ks) {
            const int kb = ks * 4;
            v2f a = *(const v2f*)(arow + kb);        // 8B-aligned ds_load_b64
            const int t0  = kb + khalf;
            const int ft0 = (fbin * t0) & 255;       // exact range reduction
            const int ft1 = (fbin * (t0 + 1)) & 255;
            v2f bc, bs;
            bc.x = __cosf((float)ft0 * PI_OVER_128);
            bc.y = __cosf((float)ft1 * PI_OVER_128);
            bs.x = __sinf((float)ft0 * PI_OVER_128);
            bs.y = __sinf((float)ft1 * PI_OVER_128);
            accC = __builtin_amdgcn_wmma_f32_16x16x4_f32(
                false, a, false, bc, (short)0, accC, false, false);
            accS = __builtin_amdgcn_wmma_f32_16x16x4_f32(
                false, a, false, bs, (short)0, accS, false, false);
        }
        // |spec|^2, summed over this tile's 16 segment rows (padded rows are 0)
        float colsum = 0.f;
        for (int v = 0; v < 8; ++v)
            colsum += accC[v] * accC[v] + accS[v] * accS[v];
        colsum += __shfl_xor(colsum, 16, 32);        // join M=v and M=v+8 halves
        if (lane < 16)
            psdp[w * 80 + ntile * 16 + mn] += colsum; // per-wave slot: race-free
    }
    __syncthreads();

    // ---- stage 7: deterministic PSD reduce + first-argmax + contribution ----
    for (int j = tid; j < NBAND; j += NTHREADS) {
        float s = 0.f;
        for (int ww = 0; ww < 8; ++ww) s += psdp[ww * 80 + j];
        psd[j] = s;
    }
    __syncthreads();
    if (tid == 0) {
        int jmax = 0; float best = psd[0];
        for (int j = 1; j < NBAND; ++j)
            if (psd[j] > best) { best = psd[j]; jmax = j; }  // first max, like argmax
        float rr   = 60.0f * (float)(LOBIN + jmax) * (1.0f / 256.0f);
        float temp = fabsf(rr - pre[b]);
        contrib[b] = (temp >= 5.0f) ? temp : 0.0f;   // |sel - rr|
    }
}

__global__ void bvp_rr_reduce_kernel(const float* __restrict__ contrib,
                                     float* __restrict__ out, int B)
{
    __shared__ float red[NTHREADS];
    const int tid = threadIdx.x;
    float s = 0.f;
    for (int j = tid; j < B; j += NTHREADS) s += contrib[j];
    red[tid] = s;
    __syncthreads();
    for (int off = NTHREADS / 2; off > 0; off >>= 1) {
        if (tid < off) red[tid] += red[tid + off];
        __syncthreads();
    }
    if (tid == 0) out[0] = red[0] / (float)B;
}

extern "C" void kernel_launch(void* const* d_in, const int* in_sizes, int n_in,
                              void* d_out, int out_size, void* d_ws, size_t ws_size,
                              hipStream_t stream) {
    const float* gt  = (const float*)d_in[0];   // [2048,1,9000] f32
    const float* pre = (const float*)d_in[1];   // [2048,1]      f32
    float* out = (float*)d_out;                 // scalar f32
    float* ws  = (float*)d_ws;                  // per-trace contributions
    const int B = in_sizes[1];                  // 2048

    bvp_rr_block_kernel<<<B, NTHREADS, SMEM_BYTES, stream>>>(gt, pre, ws);
    bvp_rr_reduce_kernel<<<1, NTHREADS, 0, stream>>>(ws, out, B);
}